// SolveBoxQP_833223655578
// MI455X (gfx1250) — compile-verified
//
#include <hip/hip_runtime.h>

typedef float v2f __attribute__((ext_vector_type(2)));
typedef float v8f __attribute__((ext_vector_type(8)));

#define NN    256
#define LDM   260      // padded LDS row stride (floats): conflict-free WMMA loads, 16B-aligned rows
#define NITER 50
#define RHO   1.0f
#define TPB   512      // 16 wave32 waves -> 16 x 16-row WMMA blocks
#define KB    8        // pivot-panel width: two chained rank-4 WMMAs per tile RMW

// Dynamic LDS layout (floats):
//   [0,      NN*LDM) : M, inverted in place to Minv (blocked Gauss-Jordan)
//   +NN              : rhs
//   +NN              : xv     (matvec result)
//   +4*NN, +4*NN     : CnegA, CnegB (negated column panel halves, 256x4, stride 4)
//   +4*NN, +4*NN     : RtA,   RtB   (row panel transposed halves,  256x4, stride 4)
//   +64              : Bbuf   (8x8 pivot-block inverse)
extern "C" __global__ __launch_bounds__(TPB, 1)
void boxqp_admm_kernel(const float* __restrict__ Q,
                       const float* __restrict__ p,
                       const float* __restrict__ lb,
                       const float* __restrict__ ub,
                       float* __restrict__ out)
{
    extern __shared__ float smem[];
    float* Mm    = smem;
    float* rhs   = smem + NN * LDM;
    float* xv    = rhs + NN;
    float* CnegA = xv + NN;
    float* CnegB = CnegA + 4 * NN;
    float* RtA   = CnegB + 4 * NN;
    float* RtB   = RtA + 4 * NN;
    float* Bbuf  = RtB + 4 * NN;

    const int tid  = threadIdx.x;
    const int b    = blockIdx.x;
    const int lane = tid & 31;
    const int wv   = tid >> 5;                 // wave id: 16 waves, one 16-row block each
    const int rowA = (wv << 4) + (lane & 15);  // WMMA A-operand row for this lane
    const int hk   = (lane >> 4) << 1;         // K sub-offset: 0 (lanes 0-15) / 2 (lanes 16-31)
    const float* Qb = Q + (size_t)b * NN * NN;

    // ---- Phase 1: load M = Q + rho*I into LDS (coalesced float4 loads) ----
    for (int idx = tid; idx < NN * (NN / 4); idx += TPB) {
        int row = idx >> 6;
        int c   = (idx & 63) << 2;
        float4 q = *(const float4*)(Qb + row * NN + c);
        float* d = Mm + row * LDM + c;
        d[0] = q.x; d[1] = q.y; d[2] = q.z; d[3] = q.w;
    }
    __syncthreads();
    if (tid < NN) Mm[tid * LDM + tid] += RHO;
    __syncthreads();

    // ---- Phase 2: blocked in-place Gauss-Jordan inversion (SPD, no pivoting),
    //      8-wide panels -> two chained rank-4 V_WMMA_F32_16X16X4_F32 per tile ----
    for (int K = 0; K < NN; K += KB) {
        // (1a) save negated column panel (zeroed on panel rows)
        if (tid < NN) {
            bool inK = (tid >= K) && (tid < K + KB);
            float4 cA = *(const float4*)(Mm + tid * LDM + K);
            float4 cB = *(const float4*)(Mm + tid * LDM + K + 4);
            float4 nA, nB;
            nA.x = inK ? 0.f : -cA.x;  nA.y = inK ? 0.f : -cA.y;
            nA.z = inK ? 0.f : -cA.z;  nA.w = inK ? 0.f : -cA.w;
            nB.x = inK ? 0.f : -cB.x;  nB.y = inK ? 0.f : -cB.y;
            nB.z = inK ? 0.f : -cB.z;  nB.w = inK ? 0.f : -cB.w;
            *(float4*)(CnegA + tid * 4) = nA;
            *(float4*)(CnegB + tid * 4) = nB;
        }
        // (1b) 8x8 pivot-block inverse: 8 lanes of wave 0, GJ with __shfl broadcasts
        if (wv == 0) {
            const int lr = lane;
            float ar[8];
            #pragma unroll
            for (int j = 0; j < 8; ++j)
                ar[j] = (lr < 8) ? Mm[(K + lr) * LDM + K + j] : 0.f;
            #pragma unroll
            for (int kk = 0; kk < 8; ++kk) {
                float prow[8];
                #pragma unroll
                for (int j = 0; j < 8; ++j) prow[j] = __shfl(ar[j], kk);
                float pinv = 1.0f / prow[kk];
                if (lr == kk) {
                    #pragma unroll
                    for (int j = 0; j < 8; ++j) ar[j] = (j == kk) ? pinv : ar[j] * pinv;
                } else {
                    float f = ar[kk];
                    #pragma unroll
                    for (int j = 0; j < 8; ++j) if (j != kk) ar[j] -= f * (prow[j] * pinv);
                    ar[kk] = -f * pinv;
                }
            }
            if (lr < 8) {
                #pragma unroll
                for (int j = 0; j < 8; ++j) Bbuf[lr * 8 + j] = ar[j];
            }
        }
        __syncthreads();

        // (2) row panel R = B * A[K8,:]; in place + transposed halves RtA/RtB
        {
            int j  = tid & 255;
            int r0 = (tid >> 8) << 2;          // rows {0..3} or {4..7} of the panel
            float av[8];
            #pragma unroll
            for (int m = 0; m < 8; ++m) av[m] = Mm[(K + m) * LDM + j];
            float rv[4];
            #pragma unroll
            for (int r = 0; r < 4; ++r) {
                float s = 0.f;
                #pragma unroll
                for (int m = 0; m < 8; ++m) s += Bbuf[(r0 + r) * 8 + m] * av[m];
                rv[r] = s;
            }
            __syncthreads();
            #pragma unroll
            for (int r = 0; r < 4; ++r) Mm[(K + r0 + r) * LDM + j] = rv[r];
            float* RtX = (r0 == 0) ? RtA : RtB;
            *(float4*)(RtX + j * 4) = make_float4(rv[0], rv[1], rv[2], rv[3]);
        }
        __syncthreads();

        // (3) trailing update A += CnegA*R_A + CnegB*R_B over all 16x16 tiles
        {
            v2f ca0 = *(const v2f*)(CnegA + rowA * 4 + hk);   // reused for all cj
            v2f ca1 = *(const v2f*)(CnegB + rowA * 4 + hk);
            const int rr = (wv << 4) + ((lane >> 4) << 3);    // C/D: rows g / g+8 per half-wave
            const int cl = lane & 15;
            for (int cj = 0; cj < 16; ++cj) {
                const int cc = (cj << 4) + cl;
                v2f b0 = *(const v2f*)(RtA + cc * 4 + hk);
                v2f b1 = *(const v2f*)(RtB + cc * 4 + hk);
                v8f acc;
                #pragma unroll
                for (int g = 0; g < 8; ++g) acc[g] = Mm[(rr + g) * LDM + cc];
                acc = __builtin_amdgcn_wmma_f32_16x16x4_f32(
                    false, ca0, false, b0, (short)0, acc, false, false);
                acc = __builtin_amdgcn_wmma_f32_16x16x4_f32(
                    false, ca1, false, b1, (short)0, acc, false, false);
                #pragma unroll
                for (int g = 0; g < 8; ++g) Mm[(rr + g) * LDM + cc] = acc[g];
            }
        }
        __syncthreads();

        // (4) column fix: A[i!=K8, K8] = Cneg[i,:] * B ; A[K8,K8] = B
        if (tid < NN) {
            if (tid < K || tid >= K + KB) {
                const float* cA = CnegA + tid * 4;
                const float* cB = CnegB + tid * 4;
                float o[8];
                #pragma unroll
                for (int c = 0; c < 8; ++c) {
                    float s = 0.f;
                    #pragma unroll
                    for (int r = 0; r < 4; ++r) s += cA[r] * Bbuf[r * 8 + c];
                    #pragma unroll
                    for (int r = 0; r < 4; ++r) s += cB[r] * Bbuf[(4 + r) * 8 + c];
                    o[c] = s;
                }
                *(float4*)(Mm + tid * LDM + K)     = make_float4(o[0], o[1], o[2], o[3]);
                *(float4*)(Mm + tid * LDM + K + 4) = make_float4(o[4], o[5], o[6], o[7]);
            } else {
                int r = tid - K;
                *(float4*)(Mm + tid * LDM + K)     = *(const float4*)(Bbuf + r * 8);
                *(float4*)(Mm + tid * LDM + K + 4) = *(const float4*)(Bbuf + r * 8 + 4);
            }
        }
        __syncthreads();
    }

    // ---- Phase 3: 50 ADMM iterations, Minv resident in LDS ----
    float zi = 0.f, ui = 0.f, xi = 0.f, pi = 0.f, lbi = 0.f, ubi = 0.f;
    if (tid < NN) {
        pi  = p [(size_t)b * NN + tid];
        lbi = lb[(size_t)b * NN + tid];
        ubi = ub[(size_t)b * NN + tid];
    }
    const float* arow = Mm + rowA * LDM + hk;

    for (int it = 0; it < NITER; ++it) {
        if (tid < NN) rhs[tid] = -pi + (zi - ui);    // rho == 1
        __syncthreads();

        // x = Minv * rhs via V_WMMA_F32_16X16X4_F32; B columns all replicated
        v8f acc = {};
        #pragma unroll 4
        for (int kb = 0; kb < NN; kb += 4) {
            v2f a  = *(const v2f*)(arow + kb);
            v2f bm = *(const v2f*)(rhs + kb + hk);
            acc = __builtin_amdgcn_wmma_f32_16x16x4_f32(
                false, a, false, bm, (short)0, acc, false, false);
        }
        if (lane == 0) {
            #pragma unroll
            for (int g = 0; g < 8; ++g) xv[(wv << 4) + g] = acc[g];
        }
        if (lane == 16) {
            #pragma unroll
            for (int g = 0; g < 8; ++g) xv[(wv << 4) + 8 + g] = acc[g];
        }
        __syncthreads();

        if (tid < NN) {
            xi = xv[tid];
            float t = xi + ui;
            zi = fminf(fmaxf(t, lbi), ubi);
            ui = ui + xi - zi;
        }
        __syncthreads();
    }

    if (tid < NN) out[(size_t)b * NN + tid] = xi;
}

extern "C" void kernel_launch(void* const* d_in, const int* in_sizes, int n_in,
                              void* d_out, int out_size, void* d_ws, size_t ws_size,
                              hipStream_t stream) {
    const float* Q  = (const float*)d_in[0];
    const float* p  = (const float*)d_in[1];
    const float* lb = (const float*)d_in[2];
    const float* ub = (const float*)d_in[3];
    float* out = (float*)d_out;

    const int B = in_sizes[0] / (NN * NN);
    const size_t shmem = (size_t)(NN * LDM + 2 * NN + 16 * NN + 64) * sizeof(float);

    // CDNA5: up to 320KB LDS per workgroup; raise the dynamic-LDS cap (~278KB used).
    (void)hipFuncSetAttribute((const void*)boxqp_admm_kernel,
                              hipFuncAttributeMaxDynamicSharedMemorySize,
                              (int)shmem);

    boxqp_admm_kernel<<<B, TPB, shmem, stream>>>(Q, p, lb, ub, out);
}